// MessagePassingModel_79920751444457
// MI455X (gfx1250) — compile-verified
//
#include <hip/hip_runtime.h>
#include <math.h>

// ---------------- model dimensions (fixed by the reference) ----------------
#define FDIM   32        // F
#define NLM    9         // (MAXD+1)^2
#define NCH    18        // 2 parities * 9 lm channels
#define XSTR   576       // NCH * FDIM floats per node row
#define NMOL   256       // B
#define NAT    29        // N
#define MN     7424      // B*N nodes
#define EPM    812       // N*(N-1) edges per molecule
#define EDST   28        // edges per destination atom (consecutive in dst_idx)
#define KRAD   8

typedef float v2f __attribute__((ext_vector_type(2)));
typedef float v8f __attribute__((ext_vector_type(8)));

__device__ __forceinline__ int degree_of(int lm) { return (lm == 0) ? 0 : ((lm < 4) ? 1 : 2); }

__device__ __forceinline__ int parity_p3(int p1, int l1, int p2, int l2, int l3) {
  int s1 = (p1 ? -1 : 1) * ((l1 & 1) ? -1 : 1);
  int s2 = (p2 ? -1 : 1) * ((l2 & 1) ? -1 : 1);
  int t  = (l3 & 1) ? -1 : 1;
  return (s1 * s2 == t) ? 0 : 1;
}

// ---------------- Clebsch-Gordan (real SH) generation on device ------------
__device__ double dfact(int n) { double v = 1.0; for (int i = 2; i <= n; ++i) v *= (double)i; return v; }

__device__ double cg_complex(int j1, int m1, int j2, int m2, int j3, int m3) {
  if (m1 + m2 != m3) return 0.0;
  if (j3 < abs(j1 - j2) || j3 > j1 + j2) return 0.0;
  double pref = sqrt((2.0 * j3 + 1.0) * dfact(j3 + j1 - j2) * dfact(j3 - j1 + j2) *
                     dfact(j1 + j2 - j3) / dfact(j1 + j2 + j3 + 1));
  pref *= sqrt(dfact(j3 + m3) * dfact(j3 - m3) * dfact(j1 - m1) * dfact(j1 + m1) *
               dfact(j2 - m2) * dfact(j2 + m2));
  double s = 0.0;
  for (int k = 0; k <= j1 + j2 + j3; ++k) {
    int d0 = k, d1 = j1 + j2 - j3 - k, d2 = j1 - m1 - k, d3 = j2 + m2 - k;
    int d4 = j3 - j2 + m1 + k, d5 = j3 - j1 - m2 + k;
    if (d0 < 0 || d1 < 0 || d2 < 0 || d3 < 0 || d4 < 0 || d5 < 0) continue;
    double den = dfact(d0) * dfact(d1) * dfact(d2) * dfact(d3) * dfact(d4) * dfact(d5);
    s += ((k & 1) ? -1.0 : 1.0) / den;
  }
  return pref * s;
}

// U(l)[i][j], real-SH unitary, complex value returned in (re, im)
__device__ void uval(int l, int i, int j, double& re, double& im) {
  re = 0.0; im = 0.0;
  int mi = i - l, mj = j - l;
  const double is2 = 0.70710678118654752440;
  if (mi == 0) { if (mj == 0) re = 1.0; return; }
  if (mi > 0) {
    if (mj == mi)       re = (mi & 1) ? -is2 : is2;
    else if (mj == -mi) re = is2;
    return;
  }
  int m = -mi;
  if (mj == mi)      im = is2;                         // 1j/sqrt2
  else if (mj == m)  im = -((m & 1) ? -is2 : is2);     // -1j*(-1)^m/sqrt2
}

// cg: 27 triples * 125 dense floats; nz lists: count + packed(a,b,c) + coef
__global__ void cg_init_kernel(float* __restrict__ cg, int* __restrict__ nzc,
                               int* __restrict__ nza, float* __restrict__ nzf) {
  int tid = threadIdx.x;
  if (tid >= 27) return;
  int l1 = tid / 9, l2 = (tid / 3) % 3, l3 = tid % 3;
  if (l3 < abs(l1 - l2) || l3 > l1 + l2) { nzc[tid] = 0; return; }
  int n = 0;
  for (int a = 0; a < 2 * l1 + 1; ++a)
    for (int b = 0; b < 2 * l2 + 1; ++b)
      for (int c = 0; c < 2 * l3 + 1; ++c) {
        double sr = 0.0, si = 0.0;
        for (int xx = 0; xx < 2 * l1 + 1; ++xx)
          for (int yy = 0; yy < 2 * l2 + 1; ++yy)
            for (int zz = 0; zz < 2 * l3 + 1; ++zz) {
              double Cv = cg_complex(l1, xx - l1, l2, yy - l2, l3, zz - l3);
              if (Cv == 0.0) continue;
              double u1r, u1i, u2r, u2i, u3r, u3i;
              uval(l1, a, xx, u1r, u1i);
              uval(l2, b, yy, u2r, u2i);
              uval(l3, c, zz, u3r, u3i);   // conj applied below
              double pr = u1r * u2r - u1i * u2i;
              double pi = u1r * u2i + u1i * u2r;
              double qr = pr * u3r + pi * u3i;
              double qi = pi * u3r - pr * u3i;
              sr += qr * Cv; si += qi * Cv;
            }
        float v = (float)(sr + si);
        cg[tid * 125 + (a * 5 + b) * 5 + c] = v;
        if (fabsf(v) > 1e-7f) { nza[tid * 128 + n] = a | (b << 4) | (c << 8); nzf[tid * 128 + n] = v; ++n; }
      }
  nzc[tid] = n;
}

// ---------------- initialization kernels ----------------------------------
__global__ __launch_bounds__(256) void init_x_kernel(const int* __restrict__ Z,
                                                     const float* __restrict__ embed,
                                                     float* __restrict__ x) {
  int gid = blockIdx.x * 256 + threadIdx.x;
  int m = gid >> 5, f = gid & 31;
  float* xr = x + (size_t)m * XSTR;
  xr[f] = embed[Z[m] * FDIM + f];
#pragma unroll
  for (int ch = 1; ch < NCH; ++ch) xr[ch * FDIM + f] = 0.f;
}

__global__ __launch_bounds__(256) void init_xef_kernel(const float* __restrict__ Ef,
                                                       float* __restrict__ xef) {
  int gid = blockIdx.x * 256 + threadIdx.x;
  int m = gid >> 5, f = gid & 31;
  int mol = m / NAT;
  float v0 = 0.f, v1 = Ef[mol * 3 + 0], v2 = Ef[mol * 3 + 1], v3 = Ef[mol * 3 + 2];
  float* r = xef + (size_t)m * XSTR;
#pragma unroll
  for (int p = 0; p < 2; ++p) {
    r[(p * NLM + 0) * FDIM + f] = v0;
    r[(p * NLM + 1) * FDIM + f] = v1;
    r[(p * NLM + 2) * FDIM + f] = v2;
    r[(p * NLM + 3) * FDIM + f] = v3;
#pragma unroll
    for (int a = 4; a < NLM; ++a) r[(p * NLM + a) * FDIM + f] = 0.f;
  }
}

// ---------------- message passing (basis recomputed in-register) -----------
// One wave = one destination atom (its 28 incoming edges are consecutive),
// so segment_sum needs no atomics. Lane = feature f.
__global__ __launch_bounds__(256)
void message_kernel(const float* __restrict__ x, float* __restrict__ y,
                    const float* __restrict__ pos,
                    const int* __restrict__ dsti, const int* __restrict__ srci,
                    const float* __restrict__ W,
                    const int* __restrict__ nzc, const int* __restrict__ nza,
                    const float* __restrict__ nzf,
                    int P1, int L1, int Lout) {
  __shared__ float sB[8][72];            // per-wave edge basis (9 lm x 8 radial)
  __shared__ float sWB[8][5 * FDIM];     // per-wave basis*W per path
  __shared__ float sAcc[8][NCH * FDIM];  // per-wave output accumulator
  int wave = threadIdx.x >> 5, lane = threadIdx.x & 31;
  int m = blockIdx.x * 8 + wave;
  int mol = m / NAT, aloc = m - mol * NAT;
#pragma unroll
  for (int ch = 0; ch < NCH; ++ch) sAcc[wave][ch * FDIM + lane] = 0.f;

  for (int t = 0; t < EDST; ++t) {
    int loc = aloc * EDST + t;
    int d = mol * NAT + dsti[loc];       // == m by construction
    int s = mol * NAT + srci[loc];
    float dx = pos[3 * s + 0] - pos[3 * d + 0];
    float dy = pos[3 * s + 1] - pos[3 * d + 1];
    float dz = pos[3 * s + 2] - pos[3 * d + 2];
    float r = sqrtf(dx * dx + dy * dy + dz * dz);
    float inv = 1.f / (r + 1e-12f);
    float ux = dx * inv, uy = dy * inv, uz = dz * inv;
    float tt = 1.f / (1.f + r);
    float rc = r * 0.2f;  // r / CUT(=5)
    float fc = 0.f;
    if (rc < 1.f) { float dd = fmaxf(1.f - rc * rc, 1e-9f); fc = expf(1.f - 1.f / dd); }
    // stage basis for this edge into LDS (lanes cover 72 entries)
    for (int idx = lane; idx < 72; idx += 32) {
      int a = idx >> 3, k = idx & 7;
      float sh;
      switch (a) {
        case 0: sh = 1.f; break;
        case 1: sh = uy; break;
        case 2: sh = uz; break;
        case 3: sh = ux; break;
        case 4: sh = 1.7320508075688772f * ux * uy; break;
        case 5: sh = 1.7320508075688772f * uy * uz; break;
        case 6: sh = 0.5f * (3.f * uz * uz - 1.f); break;
        case 7: sh = 1.7320508075688772f * ux * uz; break;
        default: sh = 0.8660254037844386f * (ux * ux - uy * uy); break;
      }
      const float binom[8] = {1.f, 7.f, 21.f, 35.f, 35.f, 21.f, 7.f, 1.f};
      float Rk = binom[k] * powf(tt, (float)k) * powf(1.f - tt, (float)(7 - k));
      sB[wave][idx] = sh * Rk * fc;
    }
    __builtin_amdgcn_wave_barrier();
    asm volatile("s_wait_dscnt 0" ::: "memory");

    const float* xs = x + (size_t)s * XSTR;
    int path = 0;
    for (int p1 = 0; p1 < P1; ++p1)
      for (int l1 = 0; l1 <= L1; ++l1)
        for (int l2 = 0; l2 <= 2; ++l2) {
          int lo = abs(l1 - l2), hi = min(l1 + l2, Lout);
          for (int l3 = lo; l3 <= hi; ++l3) {
            int p3 = parity_p3(p1, l1, 0, l2, l3);
            int nb = 2 * l2 + 1, sl2 = l2 * l2;
            for (int b = 0; b < nb; ++b) {
              float acc = 0.f;
#pragma unroll
              for (int k = 0; k < KRAD; ++k)
                acc += sB[wave][(sl2 + b) * 8 + k] * W[((size_t)path * KRAD + k) * FDIM + lane];
              sWB[wave][b * FDIM + lane] = acc;
            }
            int trip = (l1 * 3 + l2) * 3 + l3;
            int n = nzc[trip];
            int sl1 = l1 * l1, sl3 = l3 * l3;
            for (int j = 0; j < n; ++j) {
              int pk = nza[trip * 128 + j];
              float coef = nzf[trip * 128 + j];
              int a = pk & 15, b = (pk >> 4) & 15, c = (pk >> 8) & 15;
              float xv = xs[(p1 * NLM + sl1 + a) * FDIM + lane];
              sAcc[wave][(p3 * NLM + sl3 + c) * FDIM + lane] += coef * xv * sWB[wave][b * FDIM + lane];
            }
            ++path;
          }
        }
    __builtin_amdgcn_wave_barrier();
    asm volatile("s_wait_dscnt 0" ::: "memory");
  }
  float* yo = y + (size_t)m * XSTR;
#pragma unroll
  for (int ch = 0; ch < NCH; ++ch) yo[ch * FDIM + lane] = sAcc[wave][ch * FDIM + lane];
}

// ---------------- equivariant dense via V_WMMA_F32_16X16X4_F32 -------------
// One wave: 16 node-rows of one (parity,lm) channel  x  32x32 weight -> 16x32
__global__ __launch_bounds__(256)
void edense_kernel(const float* __restrict__ x, const float* __restrict__ Kd,
                   const float* __restrict__ bias, float* __restrict__ y) {
  int wave = threadIdx.x >> 5, lane = threadIdx.x & 31;
  int task = blockIdx.x * 8 + wave;
  int group = task % NCH;
  int mtile = task / NCH;
  int p = group / NLM, lm = group - p * NLM;
  int deg = degree_of(lm);
  int half = lane >> 4, lrow = lane & 15;
  int m0 = mtile * 16;
  const float* A  = x + ((size_t)(m0 + lrow) * NCH + group) * FDIM;
  const float* Bm = Kd + (size_t)(p * 3 + deg) * FDIM * FDIM;
  v8f c0 = {0.f, 0.f, 0.f, 0.f, 0.f, 0.f, 0.f, 0.f};
  v8f c1 = {0.f, 0.f, 0.f, 0.f, 0.f, 0.f, 0.f, 0.f};
#pragma unroll
  for (int kc = 0; kc < 8; ++kc) {
    int k = kc * 4 + half * 2;
    v2f a = *reinterpret_cast<const v2f*>(A + k);
    v2f b0, b1;
    b0.x = Bm[(k + 0) * FDIM + lrow];
    b0.y = Bm[(k + 1) * FDIM + lrow];
    b1.x = Bm[(k + 0) * FDIM + 16 + lrow];
    b1.y = Bm[(k + 1) * FDIM + 16 + lrow];
    c0 = __builtin_amdgcn_wmma_f32_16x16x4_f32(false, a, false, b0, (short)0, c0, false, false);
    c1 = __builtin_amdgcn_wmma_f32_16x16x4_f32(false, a, false, b1, (short)0, c1, false, false);
  }
#pragma unroll
  for (int j = 0; j < 8; ++j) {
    int row = m0 + j + half * 8;
    float v0 = c0[j], v1 = c1[j];
    if (bias != nullptr && group == 0) { v0 += bias[lrow]; v1 += bias[16 + lrow]; }
    float* yr = y + ((size_t)row * NCH + group) * FDIM;
    yr[lrow] = v0;
    yr[16 + lrow] = v1;
  }
}

// head MLP layer: (M x 32) @ (32 x 32) + b, then SiLU (also WMMA fp32)
__global__ __launch_bounds__(256)
void dense_head_kernel(const float* __restrict__ hin, int in_stride,
                       const float* __restrict__ Wm, const float* __restrict__ bvec,
                       float* __restrict__ hout) {
  int wave = threadIdx.x >> 5, lane = threadIdx.x & 31;
  int mtile = blockIdx.x * 8 + wave;
  int half = lane >> 4, lrow = lane & 15;
  int m0 = mtile * 16;
  const float* A = hin + (size_t)(m0 + lrow) * in_stride;
  v8f c0 = {0.f, 0.f, 0.f, 0.f, 0.f, 0.f, 0.f, 0.f};
  v8f c1 = {0.f, 0.f, 0.f, 0.f, 0.f, 0.f, 0.f, 0.f};
#pragma unroll
  for (int kc = 0; kc < 8; ++kc) {
    int k = kc * 4 + half * 2;
    v2f a = *reinterpret_cast<const v2f*>(A + k);
    v2f b0, b1;
    b0.x = Wm[(k + 0) * FDIM + lrow];
    b0.y = Wm[(k + 1) * FDIM + lrow];
    b1.x = Wm[(k + 0) * FDIM + 16 + lrow];
    b1.y = Wm[(k + 1) * FDIM + 16 + lrow];
    c0 = __builtin_amdgcn_wmma_f32_16x16x4_f32(false, a, false, b0, (short)0, c0, false, false);
    c1 = __builtin_amdgcn_wmma_f32_16x16x4_f32(false, a, false, b1, (short)0, c1, false, false);
  }
#pragma unroll
  for (int j = 0; j < 8; ++j) {
    int row = m0 + j + half * 8;
    float v0 = c0[j] + bvec[lrow];
    float v1 = c1[j] + bvec[16 + lrow];
    v0 = v0 / (1.f + expf(-v0));
    v1 = v1 / (1.f + expf(-v1));
    hout[(size_t)row * FDIM + lrow] = v0;
    hout[(size_t)row * FDIM + 16 + lrow] = v1;
  }
}

// ---------------- tensor product (per (node, feature) thread) --------------
__global__ __launch_bounds__(256)
void tprod_kernel(const float* __restrict__ x, const float* __restrict__ yy,
                  const float* __restrict__ W,
                  const int* __restrict__ nzc, const int* __restrict__ nza,
                  const float* __restrict__ nzf,
                  float* __restrict__ out, int P2, int L2) {
  __shared__ float sOut[NCH * 256];
  int tid = threadIdx.x;
  int gid = blockIdx.x * 256 + tid;
  int m = gid >> 5, f = gid & 31;
#pragma unroll
  for (int ch = 0; ch < NCH; ++ch) sOut[ch * 256 + tid] = 0.f;
  const float* xr = x + (size_t)m * XSTR;
  const float* yr = yy + (size_t)m * XSTR;
  int path = 0;
  for (int p1 = 0; p1 < 2; ++p1)
    for (int l1 = 0; l1 <= 2; ++l1)
      for (int p2 = 0; p2 < P2; ++p2)
        for (int l2 = 0; l2 <= L2; ++l2) {
          int lo = abs(l1 - l2), hi = min(l1 + l2, 2);
          for (int l3 = lo; l3 <= hi; ++l3) {
            int p3 = parity_p3(p1, l1, p2, l2, l3);
            float wv = W[(size_t)path * FDIM + f];
            int trip = (l1 * 3 + l2) * 3 + l3;
            int n = nzc[trip];
            int sl1 = l1 * l1, sl2 = l2 * l2, sl3 = l3 * l3;
            for (int j = 0; j < n; ++j) {
              int pk = nza[trip * 128 + j];
              float coef = nzf[trip * 128 + j];
              int a = pk & 15, b = (pk >> 4) & 15, c = (pk >> 8) & 15;
              float t = xr[(p1 * NLM + sl1 + a) * FDIM + f] * yr[(p2 * NLM + sl2 + b) * FDIM + f];
              sOut[(p3 * NLM + sl3 + c) * 256 + tid] += wv * coef * t;
            }
            ++path;
          }
        }
  float* orow = out + (size_t)m * XSTR;
#pragma unroll
  for (int ch = 0; ch < NCH; ++ch) orow[ch * FDIM + f] = sOut[ch * 256 + tid];
}

// ---------------- elementwise helpers --------------------------------------
__global__ __launch_bounds__(256) void add_kernel(float* __restrict__ a, const float* __restrict__ b) {
  size_t i = (size_t)blockIdx.x * 256 + threadIdx.x;
  a[i] += b[i];
}

__global__ __launch_bounds__(256) void esilu_kernel(float* __restrict__ x) {
  int gid = blockIdx.x * 256 + threadIdx.x;
  int m = gid >> 5, f = gid & 31;
  float* xr = x + (size_t)m * XSTR;
  float s = xr[f];
  float sig = 1.f / (1.f + expf(-s));
  float gate = sig * (1.f + s * (1.f - sig));
#pragma unroll
  for (int ch = 1; ch < NCH; ++ch) xr[ch * FDIM + f] *= gate;
  xr[f] = s * sig;
}

// energy: dot with wout, add element bias, sum 29 atoms per molecule
__global__ __launch_bounds__(256)
void reduce_kernel(const float* __restrict__ h, const float* __restrict__ wout,
                   const float* __restrict__ eb, const int* __restrict__ Z,
                   float* __restrict__ out) {
  int wave = threadIdx.x >> 5, lane = threadIdx.x & 31;
  int mol = blockIdx.x * 8 + wave;
  float w = wout[lane];
  float acc = 0.f;
  for (int a = 0; a < NAT; ++a) acc += h[(size_t)(mol * NAT + a) * FDIM + lane] * w;
  if (lane < NAT) acc += eb[Z[mol * NAT + lane]];
#pragma unroll
  for (int off = 16; off > 0; off >>= 1) acc += __shfl_xor(acc, off, 32);
  if (lane == 0) out[mol] = acc;
}

// ---------------- host orchestration ---------------------------------------
extern "C" void kernel_launch(void* const* d_in, const int* in_sizes, int n_in,
                              void* d_out, int out_size, void* d_ws, size_t ws_size,
                              hipStream_t stream) {
  (void)in_sizes; (void)n_in; (void)out_size;
  const int*   Z     = (const int*)d_in[0];
  const float* pos   = (const float*)d_in[1];
  const float* Ef    = (const float*)d_in[2];
  const int*   dsti  = (const int*)d_in[3];
  const int*   srci  = (const int*)d_in[4];
  const float* embed = (const float*)d_in[5];
  const float* ebias = (const float*)d_in[6];
  const float* wout  = (const float*)d_in[7];
  // per-iteration param block: 8 + i*9 + {0:mp,1:d1k,2:d1b,3:d2k,4:d2b,5:tw,6:tdk1,7:tdk2,8:tdw}
  auto PF = [&](int i) { return (const float*)d_in[i]; };

  // workspace carve-up (256B aligned)
  size_t off = 0;
  auto carve = [&](size_t bytes) { size_t r = off; off += (bytes + 255) & ~(size_t)255; return r; };
  char* ws = (char*)d_ws;
  size_t o_cg  = carve(27 * 125 * sizeof(float));
  size_t o_nzc = carve(27 * sizeof(int));
  size_t o_nza = carve(27 * 128 * sizeof(int));
  size_t o_nzf = carve(27 * 128 * sizeof(float));
  const size_t nodeBytes = (size_t)MN * XSTR * sizeof(float);
  size_t o_x   = carve(nodeBytes);
  size_t o_xef = carve(nodeBytes);
  size_t o_y   = carve(nodeBytes);
  size_t o_t1  = carve(nodeBytes);
  size_t o_t2  = carve(nodeBytes);
  size_t o_t3  = carve(nodeBytes);
  size_t o_h1  = carve((size_t)MN * FDIM * sizeof(float));
  size_t o_h2  = carve((size_t)MN * FDIM * sizeof(float));
  if (off > ws_size) return;  // workspace too small: no-op (deterministic)

  float* cg  = (float*)(ws + o_cg);
  int*   nzc = (int*)(ws + o_nzc);
  int*   nza = (int*)(ws + o_nza);
  float* nzf = (float*)(ws + o_nzf);
  float* xp  = (float*)(ws + o_x);
  float* efp = (float*)(ws + o_xef);
  float* yp  = (float*)(ws + o_y);
  float* t1  = (float*)(ws + o_t1);
  float* t2  = (float*)(ws + o_t2);
  float* t3  = (float*)(ws + o_t3);
  float* h1  = (float*)(ws + o_h1);
  float* h2  = (float*)(ws + o_h2);

  const int gMF   = (MN * FDIM) / 256;       // 928
  const int gElem = (MN * XSTR) / 256;       // 16704
  const int gMsg  = MN / 8;                  // 928
  const int gDen  = ((MN / 16) * NCH) / 8;   // 1044
  const int gHead = (MN / 16) / 8;           // 58

  cg_init_kernel<<<1, 32, 0, stream>>>(cg, nzc, nza, nzf);
  init_x_kernel<<<gMF, 256, 0, stream>>>(Z, embed, xp);
  init_xef_kernel<<<gMF, 256, 0, stream>>>(Ef, efp);

  for (int i = 0; i < 3; ++i) {
    const float* mp   = PF(8 + i * 9 + 0);
    const float* d1k  = PF(8 + i * 9 + 1);
    const float* d1b  = PF(8 + i * 9 + 2);
    const float* d2k  = PF(8 + i * 9 + 3);
    const float* d2b  = PF(8 + i * 9 + 4);
    const float* tw   = PF(8 + i * 9 + 5);
    const float* tdk1 = PF(8 + i * 9 + 6);
    const float* tdk2 = PF(8 + i * 9 + 7);
    const float* tdw  = PF(8 + i * 9 + 8);
    int P1 = (i == 0) ? 1 : 2;
    int L1 = (i == 0) ? 0 : 2;
    int Lout = (i < 2) ? 2 : 0;
    int L2y = (i == 0) ? 1 : 2;

    message_kernel<<<gMsg, 256, 0, stream>>>(xp, yp, pos, dsti, srci, mp, nzc, nza, nzf, P1, L1, Lout);
    add_kernel<<<gElem, 256, 0, stream>>>(xp, yp);                       // x = eadd(x, y)
    edense_kernel<<<gDen, 256, 0, stream>>>(xp, d1k, d1b, t1);           // d1
    esilu_kernel<<<gMF, 256, 0, stream>>>(t1);
    edense_kernel<<<gDen, 256, 0, stream>>>(t1, d2k, d2b, xp);           // d2 -> x
    add_kernel<<<gElem, 256, 0, stream>>>(xp, yp);                       // x = eadd(x, y)
    esilu_kernel<<<gMF, 256, 0, stream>>>(xp);
    tprod_kernel<<<gMF, 256, 0, stream>>>(xp, efp, tw, nzc, nza, nzf, t2, 2, L2y);
    { float* tmp = efp; efp = t2; t2 = tmp; }                            // xEF <- tprod out
    add_kernel<<<gElem, 256, 0, stream>>>(xp, efp);                      // x = eadd(x, xEF)
    edense_kernel<<<gDen, 256, 0, stream>>>(xp, tdk1, nullptr, t1);
    edense_kernel<<<gDen, 256, 0, stream>>>(xp, tdk2, nullptr, t2);
    tprod_kernel<<<gMF, 256, 0, stream>>>(t1, t2, tdw, nzc, nza, nzf, t3, 2, 2);
    { float* tmp = xp; xp = t3; t3 = tmp; }                              // x <- tprod out
    add_kernel<<<gElem, 256, 0, stream>>>(xp, efp);                      // x = eadd(x, xEF)
    esilu_kernel<<<gMF, 256, 0, stream>>>(xp);
  }
  esilu_kernel<<<gMF, 256, 0, stream>>>(xp);

  // head MLP: first layer reads scalar channel of x (row stride = XSTR)
  dense_head_kernel<<<gHead, 256, 0, stream>>>(xp, XSTR, PF(35), PF(36), h1);
  dense_head_kernel<<<gHead, 256, 0, stream>>>(h1, FDIM, PF(37), PF(38), h2);
  dense_head_kernel<<<gHead, 256, 0, stream>>>(h2, FDIM, PF(39), PF(40), h1);
  reduce_kernel<<<NMOL / 8, 256, 0, stream>>>(h1, wout, ebias, Z, (float*)d_out);
}